// DPT_52845277610695
// MI455X (gfx1250) — compile-verified
//
#include <hip/hip_runtime.h>
#include <hip/hip_bf16.h>

#define BB 2
#define MM 8
#define LL 32
#define KKC 4
#define EE 64
#define VV 1024
#define L2V 64
#define K2V 8
#define TMAX 31
#define ROWS_MAX 48              // ceil((8 + TMAX)/16)*16
#define LOG10F 2.302585092994046f
#define EPSF 1e-8f
#define LRI 1e-3f

typedef float v2f __attribute__((ext_vector_type(2)));
typedef float v8f __attribute__((ext_vector_type(8)));

__device__ __forceinline__ int clamp_t(const int* dt) {
    int t = dt[0];
    if (t < 0) t = 0;
    if (t > TMAX) t = TMAX;
    return t;
}

// ---------------------------------------------------------------------------
// K1: per (b,m) compute U[k] = nie[L+t] @ W_k  (8 vecs)  and
//     Wv[j] = nie[par[j]] @ W_{park[j]}        (64 vecs)
// ---------------------------------------------------------------------------
__global__ void k_uw(const float* __restrict__ nie, const int* __restrict__ np,
                     const int* __restrict__ npk, const float* __restrict__ wk,
                     const int* __restrict__ dt, float* __restrict__ U,
                     float* __restrict__ Wv) {
    const int bm = blockIdx.x;
    const int t = clamp_t(dt);
    __shared__ float s_nie[L2V * EE];
    const float* niebm = nie + (size_t)bm * L2V * EE;
    for (int i = threadIdx.x; i < L2V * EE; i += blockDim.x) s_nie[i] = niebm[i];
    __syncthreads();
    const int* npbm = np + bm * L2V;
    const int* npkbm = npk + bm * L2V;
    for (int oi = threadIdx.x; oi < (K2V + L2V) * EE; oi += blockDim.x) {
        int v = oi >> 6, f = oi & 63;
        int r, c;
        if (v < K2V) { r = LL + t; c = v; }
        else { int jj = v - K2V; r = npbm[jj] & (L2V - 1); c = npkbm[jj] & (K2V - 1); }
        const float* W = wk + (size_t)c * EE * EE + f;   // W[c][e][f], stride EE over e
        const float* xr = s_nie + r * EE;
        float acc = 0.f;
        #pragma unroll 8
        for (int e = 0; e < EE; ++e) acc = fmaf(xr[e], W[(size_t)e * EE], acc);
        if (v < K2V) U[((size_t)bm * K2V + v) * EE + f] = acc;
        else         Wv[((size_t)bm * L2V + (v - K2V)) * EE + f] = acc;
    }
}

// ---------------------------------------------------------------------------
// K2: logits = X @ emb^T via V_WMMA_F32_16X16X4_F32.
//     X rows: [0..7]=u_k, [8..8+t-1]=w_i.  One block = (row-tile rt, bm).
//     8 waves x 8 col-tiles x 16 K-steps of K=4 over E=64.
// ---------------------------------------------------------------------------
__global__ void k_logits(const float* __restrict__ U, const float* __restrict__ Wv,
                         const float* __restrict__ emb, const int* __restrict__ dt,
                         float* __restrict__ LOG) {
    const int rt = blockIdx.x;       // 0..2
    const int bm = blockIdx.y;
    const int t = clamp_t(dt);
    const int nrows = K2V + t;
    if (rt * 16 >= nrows) return;    // block-uniform

    const int XS = EE + 1;           // padded stride (bank-conflict free)
    __shared__ float sX[16 * (EE + 1)];
    for (int i = threadIdx.x; i < 16 * EE; i += blockDim.x) {
        int r = i >> 6, cc = i & 63;
        int rg = rt * 16 + r;
        float val = 0.f;
        if (rg < nrows) {
            if (rg < K2V) val = U[((size_t)bm * K2V + rg) * EE + cc];
            else          val = Wv[((size_t)bm * L2V + (rg - K2V)) * EE + cc];
        }
        sX[r * XS + cc] = val;
    }
    __syncthreads();

    const int wid  = threadIdx.x >> 5;
    const int lane = threadIdx.x & 31;
    const int half = lane >> 4;      // which K pair (A) / which K pair (B)
    const int lrow = lane & 15;      // M index (A) / N index (B)

    float* outb = LOG + ((size_t)bm * ROWS_MAX + rt * 16) * VV;
    for (int tt = 0; tt < 8; ++tt) {
        const int cb = wid * 128 + tt * 16;        // vocab column base
        const float* ebase = emb + (size_t)(cb + lrow) * EE;
        v8f c = {0.f, 0.f, 0.f, 0.f, 0.f, 0.f, 0.f, 0.f};
        #pragma unroll
        for (int s = 0; s < 16; ++s) {
            const int k0 = 4 * s + 2 * half;
            v2f a; a.x = sX[lrow * XS + k0]; a.y = sX[lrow * XS + k0 + 1];
            v2f b; b.x = ebase[k0];          b.y = ebase[k0 + 1];
            c = __builtin_amdgcn_wmma_f32_16x16x4_f32(false, a, false, b,
                                                      (short)0, c, false, false);
        }
        #pragma unroll
        for (int r = 0; r < 8; ++r) {
            int row = r + 8 * half;                 // C/D layout: VGPR r -> M=r / r+8
            outb[(size_t)row * VV + cb + lrow] = c[r];
        }
    }
}

// ---------------------------------------------------------------------------
// K3: per (b,m): lse per row, then LEu[k][i] and beta[i], sum(beta)
// ---------------------------------------------------------------------------
__global__ void k_lse(const float* __restrict__ LOG, const int* __restrict__ tok,
                      const int* __restrict__ dt, float* __restrict__ LEU,
                      float* __restrict__ BETA, float* __restrict__ SUMB) {
    const int bm = blockIdx.x, b = bm / MM;
    const int t = clamp_t(dt);
    const int nrows = K2V + t;
    __shared__ float red[256];
    __shared__ float s_lse[ROWS_MAX];
    const float* Lg = LOG + (size_t)bm * ROWS_MAX * VV;

    for (int row = 0; row < nrows; ++row) {
        const float* lr = Lg + (size_t)row * VV;
        float mx = -3.0e38f;
        for (int v = threadIdx.x; v < VV; v += 256) mx = fmaxf(mx, lr[v]);
        red[threadIdx.x] = mx; __syncthreads();
        for (int s = 128; s > 0; s >>= 1) {
            if (threadIdx.x < s) red[threadIdx.x] = fmaxf(red[threadIdx.x], red[threadIdx.x + s]);
            __syncthreads();
        }
        mx = red[0]; __syncthreads();
        float se = 0.f;
        for (int v = threadIdx.x; v < VV; v += 256) se += __expf(lr[v] - mx);
        red[threadIdx.x] = se; __syncthreads();
        for (int s = 128; s > 0; s >>= 1) {
            if (threadIdx.x < s) red[threadIdx.x] += red[threadIdx.x + s];
            __syncthreads();
        }
        if (threadIdx.x == 0) s_lse[row] = mx + __logf(red[0]);
        __syncthreads();
    }
    for (int p = threadIdx.x; p < K2V * (t + 1); p += 256) {
        int k = p / (t + 1), i = p % (t + 1);
        int tk = tok[b * LL + i] & (VV - 1);
        LEU[((size_t)bm * K2V + k) * 32 + i] = Lg[(size_t)k * VV + tk] - s_lse[k];
    }
    for (int i = threadIdx.x; i < t; i += 256) {
        int tk = tok[b * LL + i] & (VV - 1);
        BETA[(size_t)bm * 32 + i] = Lg[(size_t)(K2V + i) * VV + tk] - s_lse[K2V + i];
    }
    if (threadIdx.x == 0) {
        float sb = 0.f;
        for (int i = 0; i < t; ++i)
            sb += Lg[(size_t)(K2V + i) * VV + (tok[b * LL + i] & (VV - 1))] - s_lse[K2V + i];
        SUMB[bm] = sb;
    }
}

// ---------------------------------------------------------------------------
// K4: lp[b,m,k,j] for all 8192 proposals.  Streams the noise slab i in [L,L+t].
// ---------------------------------------------------------------------------
__global__ void k_lp(const float* __restrict__ nie, const float* __restrict__ noise,
                     const float* __restrict__ U, const float* __restrict__ Wv,
                     const float* __restrict__ LEU, const float* __restrict__ BETA,
                     const float* __restrict__ SUMB, const float* __restrict__ lpg_in,
                     const int* __restrict__ dt, float* __restrict__ LP) {
    const int k = blockIdx.x, bm = blockIdx.y;
    const int t = clamp_t(dt);
    const int wid = threadIdx.x >> 5, lane = threadIdx.x & 31;
    const float* niebm = nie + (size_t)bm * L2V * EE;
    const float nmask = (t > 0) ? (float)(2 * t - 1) : 1.0f;
    const float logtot = __logf((float)K2V * (nmask + (float)L2V * EPSF));
    const float cst = -(float)((t + 1) * EE) * LOG10F;
    const float sumb = SUMB[bm];
    const float lpg = lpg_in[bm];
    const int k4 = (k + KKC) % K2V;

    for (int j = wid; j < L2V; j += 8) {
        float acc = 0.f;
        for (int s = 0; s <= t; ++s) {
            const int i = LL + s;
            const float* wv;
            if (s == 0)            wv = nullptr;
            else if (s == t)       wv = Wv + ((size_t)bm * L2V + j) * EE;
            else if (j == LL + s)  wv = U + ((size_t)bm * K2V + k4) * EE;
            else                   wv = Wv + ((size_t)bm * L2V + (LL + s)) * EE;
            const float* nz = noise + ((((size_t)bm * L2V + i) * K2V + k) * L2V + j) * EE;
            __builtin_prefetch(nz + EE, 0, 1);     // global_prefetch_b8 next line
            const float* xr = niebm + (size_t)i * EE;
            for (int e = lane; e < EE; e += 32) {
                float n = nz[e];
                float w = wv ? wv[e] : 0.f;
                float d = xr[e] + n - w;
                acc += -0.5f * d * d + 50.f * n * n;
            }
        }
        for (int off = 16; off > 0; off >>= 1) acc += __shfl_down(acc, off, 32);
        if (lane == 0) {
            float lpe = LEU[((size_t)bm * K2V + k) * 32 + t] + sumb;
            if (j < t) lpe += LEU[((size_t)bm * K2V + k4) * 32 + j] - BETA[(size_t)bm * 32 + j];
            float maskv = (t > 0) ? (((j < t) || (j > LL && j < LL + t)) ? 1.f : 0.f)
                                  : ((j == 0) ? 1.f : 0.f);
            float opc = __logf(maskv + EPSF) - logtot;
            LP[(((size_t)bm * K2V) + k) * L2V + j] = acc + cst + lpe + lpg + opc;
        }
    }
}

// ---------------------------------------------------------------------------
// K5: top-M per batch (value desc, tie -> lower flat index, jax.lax.top_k order)
// ---------------------------------------------------------------------------
__global__ void k_topk(const float* __restrict__ LP, float* __restrict__ TOPV,
                       int* __restrict__ TOPI) {
    const int b = blockIdx.x;
    __shared__ float sv[MM * K2V * L2V];
    __shared__ float rv[256];
    __shared__ int ri[256];
    const float* lpb = LP + (size_t)b * MM * K2V * L2V;
    for (int i = threadIdx.x; i < MM * K2V * L2V; i += 256) sv[i] = lpb[i];
    __syncthreads();
    for (int sel = 0; sel < MM; ++sel) {
        float bv = -3.3e38f; int bi = 0x7fffffff;
        for (int i = threadIdx.x; i < MM * K2V * L2V; i += 256) {
            float v = sv[i];
            if (v > bv || (v == bv && i < bi)) { bv = v; bi = i; }
        }
        rv[threadIdx.x] = bv; ri[threadIdx.x] = bi; __syncthreads();
        for (int s = 128; s > 0; s >>= 1) {
            if (threadIdx.x < s) {
                float v2 = rv[threadIdx.x + s]; int i2 = ri[threadIdx.x + s];
                if (v2 > rv[threadIdx.x] || (v2 == rv[threadIdx.x] && i2 < ri[threadIdx.x])) {
                    rv[threadIdx.x] = v2; ri[threadIdx.x] = i2;
                }
            }
            __syncthreads();
        }
        if (threadIdx.x == 0) {
            TOPV[b * MM + sel] = rv[0];
            TOPI[b * MM + sel] = ri[0];
            sv[ri[0]] = -3.2e38f;   // exclude
        }
        __syncthreads();
    }
}

// ---------------------------------------------------------------------------
// K6: per selected beam: inner-SGD gradient, node_ie_next, and scalar/row outputs
// ---------------------------------------------------------------------------
__global__ void k_final(const float* __restrict__ nie, const float* __restrict__ noise,
                        const float* __restrict__ U, const float* __restrict__ Wv,
                        const float* __restrict__ wk, const float* __restrict__ emb,
                        const int* __restrict__ tok, const int* __restrict__ np,
                        const int* __restrict__ npk, const float* __restrict__ lpg_in,
                        const float* __restrict__ TOPV, const int* __restrict__ TOPI,
                        const int* __restrict__ dt, float* __restrict__ out) {
    const int b = blockIdx.x / MM, mo = blockIdx.x % MM;
    const int t = clamp_t(dt);
    const int idx = TOPI[b * MM + mo];
    const int m = idx / (K2V * L2V);
    const int rem = idx % (K2V * L2V);
    const int k = rem / L2V, j = rem % L2V;
    const int bm = b * MM + m;
    const int k4 = (k + KKC) % K2V;
    const int* npbm = np + bm * L2V;
    const int* npkbm = npk + bm * L2V;
    const float* niebm = nie + (size_t)bm * L2V * EE;

    __shared__ float g[L2V * EE];         // gradient accumulator
    __shared__ float R[(TMAX + 1) * EE];  // internal residuals
    __shared__ float lg[VV];
    __shared__ float red[256];
    __shared__ float gw[EE];
    __shared__ float pe[4 * EE];

    for (int i = threadIdx.x; i < L2V * EE; i += 256) g[i] = 0.f;
    __syncthreads();

    // Phase A: direct term, i in [L, L+t]
    for (int s = 0; s <= t; ++s) {
        const int i = LL + s;
        const float* wv;
        if (s == 0)            wv = nullptr;
        else if (s == t)       wv = Wv + ((size_t)bm * L2V + j) * EE;
        else if (j == LL + s)  wv = U + ((size_t)bm * K2V + k4) * EE;
        else                   wv = Wv + ((size_t)bm * L2V + (LL + s)) * EE;
        const float* nz = noise + ((((size_t)bm * L2V + i) * K2V + k) * L2V + j) * EE;
        for (int e = threadIdx.x; e < EE; e += 256) {
            float r = niebm[(size_t)i * EE + e] + nz[e] - (wv ? wv[e] : 0.f);
            R[s * EE + e] = r;
            g[i * EE + e] += r;
        }
    }
    __syncthreads();

    // Phase B: internal scatter h = -(R) @ W^T into parent rows (root s=0 blocked)
    for (int s = 1; s <= t; ++s) {
        int c, P;
        if (s == t)           { c = npkbm[j] & (K2V - 1);      P = npbm[j] & (L2V - 1); }
        else if (j == LL + s) { c = k4;                        P = LL + t; }
        else                  { c = npkbm[LL + s] & (K2V - 1); P = npbm[LL + s] & (L2V - 1); }
        if (threadIdx.x < EE) {
            const int e = threadIdx.x;
            const float* W = wk + (size_t)c * EE * EE + (size_t)e * EE;
            float acc = 0.f;
            for (int f = 0; f < EE; ++f) acc = fmaf(R[s * EE + f], W[f], acc);
            g[P * EE + e] -= acc;
        }
        __syncthreads();
    }

    // Phase C: external scatter h = (pbar - emb[tok]) @ W^T
    for (int i2 = 0; i2 <= t; ++i2) {
        const float* x; int c, P;
        if (i2 == t)      { x = U + ((size_t)bm * K2V + k) * EE;   c = k;  P = LL + t; }
        else if (j == i2) { x = U + ((size_t)bm * K2V + k4) * EE;  c = k4; P = LL + t; }
        else              { x = Wv + ((size_t)bm * L2V + i2) * EE;
                            c = npkbm[i2] & (K2V - 1); P = npbm[i2] & (L2V - 1); }
        for (int v = threadIdx.x; v < VV; v += 256) {
            const float* er = emb + (size_t)v * EE;
            float acc = 0.f;
            for (int e = 0; e < EE; ++e) acc = fmaf(x[e], er[e], acc);
            lg[v] = acc;
        }
        __syncthreads();
        float mx = -3.0e38f;
        for (int v = threadIdx.x; v < VV; v += 256) mx = fmaxf(mx, lg[v]);
        red[threadIdx.x] = mx; __syncthreads();
        for (int s = 128; s > 0; s >>= 1) {
            if (threadIdx.x < s) red[threadIdx.x] = fmaxf(red[threadIdx.x], red[threadIdx.x + s]);
            __syncthreads();
        }
        mx = red[0]; __syncthreads();
        float se = 0.f;
        for (int v = threadIdx.x; v < VV; v += 256) se += __expf(lg[v] - mx);
        red[threadIdx.x] = se; __syncthreads();
        for (int s = 128; s > 0; s >>= 1) {
            if (threadIdx.x < s) red[threadIdx.x] += red[threadIdx.x + s];
            __syncthreads();
        }
        const float Z = red[0]; __syncthreads();
        {
            const int e = threadIdx.x & 63, ch = threadIdx.x >> 6;
            float acc = 0.f;
            for (int v = ch * 256; v < ch * 256 + 256; ++v)
                acc = fmaf(__expf(lg[v] - mx), emb[(size_t)v * EE + e], acc);
            pe[ch * EE + e] = acc;
        }
        __syncthreads();
        if (threadIdx.x < EE) {
            const int e = threadIdx.x;
            float pbar = (pe[e] + pe[EE + e] + pe[2 * EE + e] + pe[3 * EE + e]) / Z;
            gw[e] = pbar - emb[(size_t)(tok[b * LL + i2] & (VV - 1)) * EE + e];
        }
        __syncthreads();
        if (threadIdx.x < EE) {
            const int e = threadIdx.x;
            const float* W = wk + (size_t)c * EE * EE + (size_t)e * EE;
            float acc = 0.f;
            for (int f = 0; f < EE; ++f) acc = fmaf(gw[f], W[f], acc);
            g[P * EE + e] += acc;
        }
        __syncthreads();
    }

    // Outputs
    const size_t O_LG  = (size_t)BB * MM;
    const size_t O_NIE = 2 * O_LG;
    const size_t O_NP  = O_NIE + (size_t)BB * MM * L2V * EE;
    const size_t O_NPK = O_NP + (size_t)BB * MM * L2V;
    float* o_nie = out + O_NIE + (size_t)(b * MM + mo) * L2V * EE;
    for (int i = threadIdx.x; i < L2V * EE; i += 256) o_nie[i] = niebm[i] - LRI * g[i];
    if (threadIdx.x == 0) {
        out[b * MM + mo] = TOPV[b * MM + mo];
        float nmask = (t > 0) ? (float)(2 * t - 1) : 1.f;
        float maskv = (t > 0) ? (((j < t) || (j > LL && j < LL + t)) ? 1.f : 0.f)
                              : ((j == 0) ? 1.f : 0.f);
        float opc = __logf(maskv + EPSF) - __logf((float)K2V * (nmask + (float)L2V * EPSF));
        out[O_LG + b * MM + mo] = lpg_in[bm] + opc;
    }
    if (threadIdx.x < L2V) {
        const int i = threadIdx.x;
        int pv, kv;
        if (i == LL + t) { pv = npbm[j]; kv = npkbm[j]; }       // row L+t wins
        else if (i == t) { pv = LL + t;  kv = k; }              // row t next
        else if (i == j) { pv = LL + t;  kv = k4; }             // diagonal last
        else             { pv = npbm[i]; kv = npkbm[i]; }
        out[O_NP  + (size_t)(b * MM + mo) * L2V + i] = (float)pv;
        out[O_NPK + (size_t)(b * MM + mo) * L2V + i] = (float)kv;
    }
}

// ---------------------------------------------------------------------------
// Workspace layout (floats)
// ---------------------------------------------------------------------------
static constexpr size_t OFF_U    = 0;
static constexpr size_t OFF_WV   = OFF_U    + (size_t)BB * MM * K2V * EE;       // 8192
static constexpr size_t OFF_LOG  = OFF_WV   + (size_t)BB * MM * L2V * EE;       // 65536
static constexpr size_t OFF_LEU  = OFF_LOG  + (size_t)BB * MM * ROWS_MAX * VV;  // 786432
static constexpr size_t OFF_BETA = OFF_LEU  + (size_t)BB * MM * K2V * 32;
static constexpr size_t OFF_SUMB = OFF_BETA + (size_t)BB * MM * 32;
static constexpr size_t OFF_LP   = OFF_SUMB + (size_t)BB * MM;
static constexpr size_t OFF_TOPV = OFF_LP   + (size_t)BB * MM * K2V * L2V;
static constexpr size_t OFF_TOPI = OFF_TOPV + (size_t)BB * MM;

extern "C" void kernel_launch(void* const* d_in, const int* in_sizes, int n_in,
                              void* d_out, int out_size, void* d_ws, size_t ws_size,
                              hipStream_t stream) {
    (void)in_sizes; (void)n_in; (void)out_size; (void)ws_size;
    const int*   tok  = (const int*)d_in[0];
    const float* lpg  = (const float*)d_in[1];
    const float* nie  = (const float*)d_in[2];
    const int*   np   = (const int*)d_in[3];
    const int*   npk  = (const int*)d_in[4];
    const float* emb  = (const float*)d_in[5];
    const float* wk   = (const float*)d_in[6];
    const float* nz   = (const float*)d_in[7];
    const int*   dt   = (const int*)d_in[8];

    float* ws   = (float*)d_ws;
    float* U    = ws + OFF_U;
    float* Wv   = ws + OFF_WV;
    float* LOG  = ws + OFF_LOG;
    float* LEU  = ws + OFF_LEU;
    float* BETA = ws + OFF_BETA;
    float* SUMB = ws + OFF_SUMB;
    float* LP   = ws + OFF_LP;
    float* TOPV = ws + OFF_TOPV;
    int*   TOPI = (int*)(ws + OFF_TOPI);
    float* out  = (float*)d_out;

    k_uw    <<<BB * MM, 256, 0, stream>>>(nie, np, npk, wk, dt, U, Wv);
    k_logits<<<dim3(ROWS_MAX / 16, BB * MM), 256, 0, stream>>>(U, Wv, emb, dt, LOG);
    k_lse   <<<BB * MM, 256, 0, stream>>>(LOG, tok, dt, LEU, BETA, SUMB);
    k_lp    <<<dim3(K2V, BB * MM), 256, 0, stream>>>(nie, nz, U, Wv, LEU, BETA, SUMB, lpg, dt, LP);
    k_topk  <<<BB, 256, 0, stream>>>(LP, TOPV, TOPI);
    k_final <<<BB * MM, 256, 0, stream>>>(nie, nz, U, Wv, wk, emb, tok, np, npk, lpg,
                                          TOPV, TOPI, dt, out);
}